// AVSegFormer_40673340293644
// MI455X (gfx1250) — compile-verified
//
#include <hip/hip_runtime.h>

typedef __attribute__((ext_vector_type(2))) float v2f;
typedef __attribute__((ext_vector_type(8))) float v8f;

#define GENE_DIM 32
#define EMBED_DIM 256
#define T_NODES 128
#define IN_DIM 35
#define K_PAD 36
#define NEG_SLOPE 0.2f
#define NCELLS 1024   // B*Q = 4*256

// workspace layout (floats):
//   [0 .. 63]            v = W_gat @ a_src (padded, v[35..63]=0)
//   [64 .. 64+1023]      per-cell active flag (1.0 / 0.0)
//   [1088 .. ]           y[cell][K_PAD]
#define WS_V     0
#define WS_FLAG  64
#define WS_Y     (64 + NCELLS)

// ---------------------------------------------------------------------------
// Kernel 0: v[k] = sum_d W[k][d] * a_src[d]   (35x256 -> 35), padded to 64
// ---------------------------------------------------------------------------
__global__ void precompute_v(const float* __restrict__ W,
                             const float* __restrict__ a_src,
                             float* __restrict__ ws) {
  int k = threadIdx.x;                    // 64 threads
  float acc = 0.f;
  if (k < IN_DIM) {
    for (int d = 0; d < EMBED_DIM; ++d)
      acc += W[k * EMBED_DIM + d] * a_src[d];
  }
  ws[WS_V + k] = (k < IN_DIM) ? acc : 0.f;
}

// ---------------------------------------------------------------------------
// Kernel 1: one block (128 threads = 4 waves) per cell.
// Build transcript feature x[35], asrc = x.v, leaky-relu, softmax over valid
// transcripts + self-loop (logit 0), reduce y = sum_t w_t * x_t.
// ---------------------------------------------------------------------------
__global__ void cell_reduce(const float* __restrict__ omics_x,     // [B,Q,T,2]
                            const float* __restrict__ centroids,   // [B,Q,2]
                            const int*   __restrict__ gene_ids,    // [B,Q,T]
                            const float* __restrict__ qv,          // [B,Q,T,1]
                            const unsigned char* __restrict__ tmask,  // [B,Q,T]
                            const unsigned char* __restrict__ qmask,  // [B,Q]
                            const float* __restrict__ gene_emb,    // [20000,32]
                            float* __restrict__ ws) {
  __shared__ float red[T_NODES];
  __shared__ float xs[K_PAD * T_NODES];   // 18 KB, transposed [k][t]

  const float* v   = ws + WS_V;
  float* flags     = ws + WS_FLAG;
  float* y         = ws + WS_Y;

  const int cell = blockIdx.x;
  const int t    = threadIdx.x;           // 0..127
  const bool valid = tmask[cell * T_NODES + t] != 0;

  // --- gather the 35-dim transcript feature ---
  float x[K_PAD];
  const float cx = centroids[cell * 2 + 0];
  const float cy = centroids[cell * 2 + 1];
  x[0] = omics_x[(cell * T_NODES + t) * 2 + 0] - cx;
  x[1] = omics_x[(cell * T_NODES + t) * 2 + 1] - cy;
  const int gid = gene_ids[cell * T_NODES + t];
#pragma unroll
  for (int g = 0; g < GENE_DIM; ++g)
    x[2 + g] = gene_emb[gid * GENE_DIM + g];
  x[34] = qv[cell * T_NODES + t];
  x[35] = 0.f;

  // --- attention logit for anchor row: leaky_relu(asrc_t), adst[anchor]=0 ---
  float asrc = 0.f;
#pragma unroll
  for (int k = 0; k < K_PAD; ++k) asrc += x[k] * v[k];
  const float l = (asrc >= 0.f) ? asrc : NEG_SLOPE * asrc;

  // --- max over valid transcripts ---
  red[t] = valid ? l : -3.0e38f;
  __syncthreads();
  for (int s = T_NODES / 2; s > 0; s >>= 1) {
    if (t < s) red[t] = fmaxf(red[t], red[t + s]);
    __syncthreads();
  }
  const float m = fmaxf(red[0], 0.f);     // include self-loop logit 0
  __syncthreads();

  // --- sum of exps over valid transcripts ---
  const float e = valid ? expf(l - m) : 0.f;
  red[t] = e;
  __syncthreads();
  for (int s = T_NODES / 2; s > 0; s >>= 1) {
    if (t < s) red[t] += red[t + s];
    __syncthreads();
  }
  const float Z = red[0] + expf(0.f - m); // + self-loop term
  const float w = e / Z;
  __syncthreads();

  // --- any-valid reduction ---
  red[t] = valid ? 1.f : 0.f;
  __syncthreads();
  for (int s = T_NODES / 2; s > 0; s >>= 1) {
    if (t < s) red[t] = fmaxf(red[t], red[t + s]);
    __syncthreads();
  }
  const float anyvalid = red[0];

  // --- y = sum_t w_t * x_t  (store transposed, then 36 threads sum) ---
#pragma unroll
  for (int k = 0; k < K_PAD; ++k) xs[k * T_NODES + t] = w * x[k];
  __syncthreads();
  if (t < K_PAD) {
    float acc = 0.f;
    for (int i = 0; i < T_NODES; ++i) acc += xs[t * T_NODES + i];
    y[cell * K_PAD + t] = acc;
  }
  if (t == 0)
    flags[cell] = ((qmask[cell] != 0) && (anyvalid > 0.5f)) ? 1.f : 0.f;
}

// ---------------------------------------------------------------------------
// Kernel 2: WMMA GEMM  Y[1024,36] @ W[36,256] -> out[1024,256], +bias, mask.
// One block (256 thr = 8 waves) per group of 16 cells; each wave owns 32
// columns (2x 16-col WMMA tiles); K loop of 9 steps of V_WMMA_F32_16X16X4_F32.
// ---------------------------------------------------------------------------
__global__ void gemm_wmma(const float* __restrict__ Wg,     // [35,256]
                          const float* __restrict__ b_gat,  // [256]
                          const float* __restrict__ ws,
                          float* __restrict__ out) {
  __shared__ float Ylds[16 * K_PAD];
  __shared__ float Flds[16];

  const float* flags = ws + WS_FLAG;
  const float* y     = ws + WS_Y;

  const int group = blockIdx.x;           // 64 groups of 16 cells
  const int tid   = threadIdx.x;          // 256 threads

  for (int i = tid; i < 16 * K_PAD; i += 256)
    Ylds[i] = y[group * 16 * K_PAD + i];
  if (tid < 16) Flds[tid] = flags[group * 16 + tid];
  __syncthreads();

  const int lane    = tid & 31;
  const int wave    = tid >> 5;           // 0..7
  const int nbase   = wave * 32;
  const int halfsel = lane >> 4;          // 0: lanes 0-15, 1: lanes 16-31
  const int lid     = lane & 15;
  const int n0 = nbase + lid;
  const int n1 = nbase + 16 + lid;

  v8f acc0 = {};
  v8f acc1 = {};
#pragma unroll
  for (int k0 = 0; k0 < K_PAD; k0 += 4) {
    const int ka = k0 + 2 * halfsel;      // this lane-half's K pair
    // A fragment: 16x4 slice of Y tile (rows = cells)
    v2f a;
    a.x = Ylds[lid * K_PAD + ka];
    a.y = Ylds[lid * K_PAD + ka + 1];
    // B fragments: 4x16 slices of W (zero-pad rows >= 35)
    v2f b0, b1;
    b0.x = (ka     < IN_DIM) ? Wg[ka       * EMBED_DIM + n0] : 0.f;
    b0.y = (ka + 1 < IN_DIM) ? Wg[(ka + 1) * EMBED_DIM + n0] : 0.f;
    b1.x = (ka     < IN_DIM) ? Wg[ka       * EMBED_DIM + n1] : 0.f;
    b1.y = (ka + 1 < IN_DIM) ? Wg[(ka + 1) * EMBED_DIM + n1] : 0.f;
    acc0 = __builtin_amdgcn_wmma_f32_16x16x4_f32(
        false, a, false, b0, (short)0, acc0, false, false);
    acc1 = __builtin_amdgcn_wmma_f32_16x16x4_f32(
        false, a, false, b1, (short)0, acc1, false, false);
  }

  // D layout: VGPR v -> row (v + 8*halfsel), col = nbase(+16) + lid
  const float bi0 = b_gat[n0];
  const float bi1 = b_gat[n1];
#pragma unroll
  for (int vv = 0; vv < 8; ++vv) {
    const int mrow = vv + 8 * halfsel;
    const int cell = group * 16 + mrow;
    const float f  = Flds[mrow];
    out[cell * EMBED_DIM + n0] = f * (acc0[vv] + bi0);
    out[cell * EMBED_DIM + n1] = f * (acc1[vv] + bi1);
  }
}

// ---------------------------------------------------------------------------
extern "C" void kernel_launch(void* const* d_in, const int* in_sizes, int n_in,
                              void* d_out, int out_size, void* d_ws, size_t ws_size,
                              hipStream_t stream) {
  const float* omics_x   = (const float*)d_in[0];
  const float* centroids = (const float*)d_in[1];
  const int*   gene_ids  = (const int*)d_in[2];
  const float* qv        = (const float*)d_in[3];
  const unsigned char* tmask = (const unsigned char*)d_in[4];
  const unsigned char* qmask = (const unsigned char*)d_in[5];
  const float* gene_emb  = (const float*)d_in[6];
  const float* Wg        = (const float*)d_in[7];
  const float* a_src     = (const float*)d_in[8];
  // d_in[9] = a_dst: unused (anchor h = 0 => adst[0] = 0)
  const float* b_gat     = (const float*)d_in[10];

  float* ws  = (float*)d_ws;
  float* out = (float*)d_out;

  precompute_v<<<1, 64, 0, stream>>>(Wg, a_src, ws);
  cell_reduce<<<NCELLS, T_NODES, 0, stream>>>(omics_x, centroids, gene_ids, qv,
                                              tmask, qmask, gene_emb, ws);
  gemm_wmma<<<NCELLS / 16, 256, 0, stream>>>(Wg, b_gat, ws, out);
}